// CoupledTauModel_69123203662072
// MI455X (gfx1250) — compile-verified
//
#include <hip/hip_runtime.h>
#include <hip/hip_bf16.h>
#include <math.h>

#define NN   8192
#define HID  128
#define BAT  8
#define RNK  5
#define NSTEPS 10
#define DTF  0.1f
#define PFDIST 4096   /* floats = 16KB prefetch-ahead on the L stream */

typedef __attribute__((ext_vector_type(2))) float v2f;
typedef __attribute__((ext_vector_type(8))) float v8f;

__device__ __forceinline__ float softplusf(float x) {
  return x > 20.f ? x : log1pf(expf(x));
}

// ---------------------------------------------------------------------------
// Tiny kernel: softplus(raw)+eps scalars -> ws
// ---------------------------------------------------------------------------
__global__ void k_scalars(const float* __restrict__ rcs, const float* __restrict__ rcf,
                          const float* __restrict__ rls, const float* __restrict__ rlf,
                          float* __restrict__ sc) {
  if (threadIdx.x == 0) {
    sc[0] = softplusf(rcs[0]) + 1e-4f;   // c_s
    sc[1] = softplusf(rcf[0]) + 1e-4f;   // c_f
    sc[2] = softplusf(rls[0]) + 1e-4f;   // l_s
    sc[3] = softplusf(rlf[0]) + 1e-4f;   // l_f
  }
}

// ---------------------------------------------------------------------------
// MLP layer 1 (encoder gate and decoder): H = relu((Xa[+Xb]) @ W + b)
// W is [NN, HID] row-major -> coalesced across h. One block per batch row.
// 4 independent accumulators to break the serial FMA chain.
// ---------------------------------------------------------------------------
__global__ void k_mlp1(const float* __restrict__ Xa, const float* __restrict__ Xb,
                       const float* __restrict__ W, const float* __restrict__ bias,
                       float* __restrict__ H) {
  const int b = blockIdx.x;
  const int h = threadIdx.x;           // 128 threads
  float acc[4] = {0.f, 0.f, 0.f, 0.f};
  if (Xb != nullptr) {
    for (int j = 0; j < NN; j += 4) {
#pragma unroll
      for (int q = 0; q < 4; ++q)
        acc[q] += (Xa[b * NN + j + q] + Xb[b * NN + j + q]) * W[(j + q) * HID + h];
    }
  } else {
    for (int j = 0; j < NN; j += 4) {
#pragma unroll
      for (int q = 0; q < 4; ++q)
        acc[q] += Xa[b * NN + j + q] * W[(j + q) * HID + h];
    }
  }
  const float a = (acc[0] + acc[1]) + (acc[2] + acc[3]) + bias[h];
  H[b * HID + h] = a > 0.f ? a : 0.f;
}

// ---------------------------------------------------------------------------
// Gate + split: gate = sigmoid(H @ W2 + b2); us = gate*x0; uf = (1-gate)*x0
// ---------------------------------------------------------------------------
__global__ void k_gate(const float* __restrict__ H, const float* __restrict__ W2,
                       const float* __restrict__ b2, const float* __restrict__ x0,
                       float* __restrict__ U0, float* __restrict__ F0) {
  __shared__ float Hs[HID];
  const int b = blockIdx.y;
  const int i = blockIdx.x * 256 + threadIdx.x;
  if (threadIdx.x < HID) Hs[threadIdx.x] = H[b * HID + threadIdx.x];
  __syncthreads();
  float acc = 0.f;
#pragma unroll 8
  for (int k = 0; k < HID; ++k) acc += Hs[k] * W2[(size_t)k * NN + i];
  acc += b2[i];
  const float g = 1.f / (1.f + expf(-acc));
  const float x = x0[b * NN + i];
  U0[b * NN + i] = g * x;
  F0[b * NN + i] = (1.f - g) * x;
}

// ---------------------------------------------------------------------------
// Low-rank coupling pre-pass per step:
//   t[0][b][r] = sum_j uf[b,j] * Ms_B[j,r]   (feeds the us-update)
//   t[1][b][r] = sum_j us[b,j] * Mf_B[j,r]   (feeds the uf-update)
// ---------------------------------------------------------------------------
__global__ void k_coup(const float* __restrict__ Uc, const float* __restrict__ Fc,
                       const float* __restrict__ MsB, const float* __restrict__ MfB,
                       float* __restrict__ tb) {
  const int sel = blockIdx.x;          // 0: from uf via Ms_B ; 1: from us via Mf_B
  const int b   = blockIdx.y;
  const float* __restrict__ src = sel ? Uc  : Fc;
  const float* __restrict__ MB  = sel ? MfB : MsB;
  __shared__ float red[256];
  float acc[RNK] = {0.f, 0.f, 0.f, 0.f, 0.f};
  for (int j = threadIdx.x; j < NN; j += 256) {
    const float v = src[b * NN + j];
#pragma unroll
    for (int r = 0; r < RNK; ++r) acc[r] += v * MB[j * RNK + r];
  }
#pragma unroll
  for (int r = 0; r < RNK; ++r) {
    red[threadIdx.x] = acc[r];
    __syncthreads();
    for (int s = 128; s > 0; s >>= 1) {
      if (threadIdx.x < (unsigned)s) red[threadIdx.x] += red[threadIdx.x + s];
      __syncthreads();
    }
    if (threadIdx.x == 0) tb[sel * (BAT * RNK) + b * RNK + r] = red[0];
    __syncthreads();
  }
}

// ---------------------------------------------------------------------------
// Main step kernel (bandwidth-bound, f32 WMMA):
//   Y[b,i] = sum_j U[b,j] * L[i,j]           (L streamed once from HBM, NT)
//   Unext[b,i] = relu(U[b,i] + dt*(lam*coup[b,i] - c*Y[b,i]))
// Per wave: D = A(16x4) x B(4x16) accumulate; M=batch (8 valid, rows 8..15
// duplicate rows 0..7, their outputs ignored), N=16 nodes, K marches over j.
// A layout  (ISA 16x4 f32): lane m<16 holds U[m, j..j+1]; lane m+16 -> j+2..j+3
// B layout  (mirrored)    : lane n<16 holds L[i0+n, j..j+1]; lane n+16 -> j+2..j+3
// ---------------------------------------------------------------------------
__global__ __launch_bounds__(128) void k_step(
    const float* __restrict__ Ucur, const float* __restrict__ Fcur,
    const float* __restrict__ Ls,   const float* __restrict__ Lf,
    const float* __restrict__ tbuf, const float* __restrict__ MsA,
    const float* __restrict__ MfA,  const float* __restrict__ sc,
    float* __restrict__ Unext,      float* __restrict__ Fnext) {
  const int sel = blockIdx.y;
  const float* __restrict__ U  = sel ? Fcur : Ucur;
  const float* __restrict__ L  = sel ? Lf   : Ls;
  const float* __restrict__ MA = sel ? MfA  : MsA;
  const float* __restrict__ tv = tbuf + sel * (BAT * RNK);
  float* __restrict__ Out = sel ? Fnext : Unext;

  const int lane  = threadIdx.x & 31;
  const int wave  = threadIdx.x >> 5;
  const int iBase = blockIdx.x * 64 + wave * 16;   // 4 waves x 16 nodes / block

  const int n16  = lane & 15;         // node index within tile (B) / batch row (A)
  const int kOff = (lane >> 4) * 2;   // K sub-offset: lanes 16..31 carry K=2,3

  const float* __restrict__ bp = L + (size_t)(iBase + n16) * NN + kOff;
  const float* __restrict__ ap = U + (size_t)(lane & 7) * NN + kOff;

  v8f acc = {};
  int j = 0;
  // Main loop: unconditional prefetch 16KB ahead (no per-iteration predicate).
  for (; j < NN - PFDIST; j += 32) {
    __builtin_prefetch(bp + j + PFDIST, 0, 0);
#pragma unroll
    for (int jj = 0; jj < 32; jj += 4) {
      v2f a = *(const v2f*)(ap + j + jj);
      v2f b = __builtin_nontemporal_load((const v2f*)(bp + j + jj));
      acc = __builtin_amdgcn_wmma_f32_16x16x4_f32(
          false, a, false, b, (short)0, acc, false, false);
    }
  }
  // Tail: last 16KB of each row, already prefetched.
  for (; j < NN; j += 32) {
#pragma unroll
    for (int jj = 0; jj < 32; jj += 4) {
      v2f a = *(const v2f*)(ap + j + jj);
      v2f b = __builtin_nontemporal_load((const v2f*)(bp + j + jj));
      acc = __builtin_amdgcn_wmma_f32_16x16x4_f32(
          false, a, false, b, (short)0, acc, false, false);
    }
  }

  // Epilogue: lanes 0..15 hold all 8 valid batch rows (acc[0..7]) for node i.
  if (lane < 16) {
    const int i = iBase + lane;
    const float c   = sc[sel];
    const float lam = sc[2 + sel];
    float ar[RNK];
#pragma unroll
    for (int r = 0; r < RNK; ++r) ar[r] = MA[i * RNK + r];
#pragma unroll
    for (int m = 0; m < BAT; ++m) {
      float coup = 0.f;
#pragma unroll
      for (int r = 0; r < RNK; ++r) coup += tv[m * RNK + r] * ar[r];
      float un = U[(size_t)m * NN + i] + DTF * (lam * coup - c * acc[m]);
      Out[(size_t)m * NN + i] = un > 0.f ? un : 0.f;
    }
  }
}

// ---------------------------------------------------------------------------
// Decoder layer 2: x1_hat = softplus(H @ D2 + db2)
// ---------------------------------------------------------------------------
__global__ void k_dec2(const float* __restrict__ H, const float* __restrict__ D2,
                       const float* __restrict__ db2, float* __restrict__ out) {
  __shared__ float Hs[HID];
  const int b = blockIdx.y;
  const int i = blockIdx.x * 256 + threadIdx.x;
  if (threadIdx.x < HID) Hs[threadIdx.x] = H[b * HID + threadIdx.x];
  __syncthreads();
  float acc = 0.f;
#pragma unroll 8
  for (int k = 0; k < HID; ++k) acc += Hs[k] * D2[(size_t)k * NN + i];
  acc += db2[i];
  out[b * NN + i] = softplusf(acc);
}

// Copy final us/uf into output slots 1 and 2.
__global__ void k_copy(const float* __restrict__ U, const float* __restrict__ F,
                       float* __restrict__ out) {
  const int idx = blockIdx.x * 256 + threadIdx.x;
  out[BAT * NN + idx]     = U[idx];
  out[2 * BAT * NN + idx] = F[idx];
}

// ---------------------------------------------------------------------------
extern "C" void kernel_launch(void* const* d_in, const int* in_sizes, int n_in,
                              void* d_out, int out_size, void* d_ws, size_t ws_size,
                              hipStream_t stream) {
  const float* x0  = (const float*)d_in[0];
  const float* Ls  = (const float*)d_in[1];
  const float* Lf  = (const float*)d_in[2];
  const float* W1  = (const float*)d_in[3];
  const float* b1  = (const float*)d_in[4];
  const float* W2  = (const float*)d_in[5];
  const float* b2  = (const float*)d_in[6];
  const float* MsA = (const float*)d_in[7];
  const float* MsB = (const float*)d_in[8];
  const float* MfA = (const float*)d_in[9];
  const float* MfB = (const float*)d_in[10];
  const float* rcs = (const float*)d_in[11];
  const float* rcf = (const float*)d_in[12];
  const float* rls = (const float*)d_in[13];
  const float* rlf = (const float*)d_in[14];
  const float* D1  = (const float*)d_in[15];
  const float* db1 = (const float*)d_in[16];
  const float* D2  = (const float*)d_in[17];
  const float* db2 = (const float*)d_in[18];

  // Workspace layout (floats): [0..3] scalars, [16..95] t, [128..1151] H,
  // then 4 ping-pong state buffers of BAT*NN each (~1.01 MB total).
  float* w  = (float*)d_ws;
  float* sc = w;
  float* tb = w + 16;
  float* H  = w + 128;
  float* U0 = w + 2048;
  float* U1 = U0 + BAT * NN;
  float* F0 = U1 + BAT * NN;
  float* F1 = F0 + BAT * NN;

  float* out = (float*)d_out;

  k_scalars<<<1, 32, 0, stream>>>(rcs, rcf, rls, rlf, sc);
  k_mlp1<<<BAT, HID, 0, stream>>>(x0, nullptr, W1, b1, H);
  k_gate<<<dim3(NN / 256, BAT), 256, 0, stream>>>(H, W2, b2, x0, U0, F0);

  float* Uc = U0; float* Un = U1; float* Fc = F0; float* Fn = F1;
  for (int s = 0; s < NSTEPS; ++s) {
    k_coup<<<dim3(2, BAT), 256, 0, stream>>>(Uc, Fc, MsB, MfB, tb);
    k_step<<<dim3(NN / 64, 2), 128, 0, stream>>>(Uc, Fc, Ls, Lf, tb, MsA, MfA, sc, Un, Fn);
    float* t0 = Uc; Uc = Un; Un = t0;
    float* t1 = Fc; Fc = Fn; Fn = t1;
  }

  k_mlp1<<<BAT, HID, 0, stream>>>(Uc, Fc, D1, db1, H);           // latent = us+uf
  k_dec2<<<dim3(NN / 256, BAT), 256, 0, stream>>>(H, D2, db2, out);
  k_copy<<<(BAT * NN) / 256, 256, 0, stream>>>(Uc, Fc, out);
}